// sparse_agg_fuse_45964740001831
// MI455X (gfx1250) — compile-verified
//
#include <hip/hip_runtime.h>
#include <math.h>

typedef float v2f __attribute__((ext_vector_type(2)));
typedef float v4f __attribute__((ext_vector_type(4)));
typedef float v8f __attribute__((ext_vector_type(8)));

#define B_   2
#define NA_  16384
#define NB_  8192
#define C_   128
#define KNN  16
#define NTILES (NA_ / 16)

// ---------------------------------------------------------------------------
// Pre-pass: pack teacher positions into [x, y, z, |t|^2] float4 (in d_ws).
// One coalesced b128 load per teacher-tile lane in the hot loop.
// ---------------------------------------------------------------------------
__global__ __launch_bounds__(256) void pack_teachers_kernel(
    const float* __restrict__ tpos, v4f* __restrict__ packed, int total)
{
    int i = blockIdx.x * blockDim.x + threadIdx.x;
    if (i < total) {
        float x = tpos[i * 3 + 0];
        float y = tpos[i * 3 + 1];
        float z = tpos[i * 3 + 2];
        v4f p; p.x = x; p.y = y; p.z = z; p.w = x * x + y * y + z * z;
        packed[i] = p;
    }
}

// ---------------------------------------------------------------------------
// Main kernel: one wave32 per 16-student tile.
//   score(t, s) = |t|^2 - 2 t.s      (row-constant |s|^2 added post-selection)
// computed as V_WMMA_F32_16X16X4_F32 with
//   A (16x4)  row t = [tx, ty, tz, |t|^2]   (VGPR0: K0/K2, VGPR1: K1/K3)
//   B (4x16)  col s = [-2sx, -2sy, -2sz, 1]
// D layout: lane n(<16) = student n, rows 0..7; lane n+16 = student n, rows 8..15.
// Each lane keeps a sorted (descending) top-16 of its half-stream; halves are
// merged by shuffles at the end (exact).
// Teacher-tile loads are software-pipelined two tiles deep so the b128 load
// latency is hidden behind the previous tiles' WMMA + scan work.
// ---------------------------------------------------------------------------
__global__ __launch_bounds__(32) void knn_agg_kernel(
    const float* __restrict__ tfeat,   // [B, NA, C]
    const v4f*  __restrict__ tpack,    // [B, NA] packed teacher [x,y,z,|t|^2]
    const float* __restrict__ spos,    // [B, NB, 3]
    float* __restrict__ out)           // [B, NB, C]
{
    const int lane    = threadIdx.x & 31;
    const int gw      = blockIdx.x;              // global wave id == tile id
    const int batch   = gw / (NB_ / 16);
    const int tile    = gw % (NB_ / 16);
    const int halfsel = lane >> 4;               // 0: lanes 0-15, 1: lanes 16-31
    const int lid     = lane & 15;

    // ---- student column operand (fixed for the whole teacher stream) ----
    const int   sIdx  = tile * 16 + lid;
    const float sx    = spos[((size_t)batch * NB_ + sIdx) * 3 + 0];
    const float sy    = spos[((size_t)batch * NB_ + sIdx) * 3 + 1];
    const float sz    = spos[((size_t)batch * NB_ + sIdx) * 3 + 2];
    const float snorm = sx * sx + sy * sy + sz * sz;

    v2f bmat;
    bmat.x = halfsel ? (-2.0f * sz) : (-2.0f * sx);   // K2 : K0
    bmat.y = halfsel ? 1.0f         : (-2.0f * sy);   // K3 : K1

    // ---- per-lane top-16 (descending: vk[0] = current worst kept) ----
    float vk[KNN];
    int   ik[KNN];
#pragma unroll
    for (int j = 0; j < KNN; ++j) { vk[j] = INFINITY; ik[j] = 0; }

    const v4f* tp = tpack + (size_t)batch * NA_;
    const int  rowoff = halfsel ? 8 : 0;

    // insert-scan of one 16-teacher tile given its packed positions
#define PROCESS_TILE(P, T)                                                    \
    {                                                                         \
        v2f amat;                                                             \
        amat.x = halfsel ? (P).z : (P).x;          /* K2 : K0 */              \
        amat.y = halfsel ? (P).w : (P).y;          /* K3 : K1 */              \
        v8f czero = {};                                                       \
        v8f c = __builtin_amdgcn_wmma_f32_16x16x4_f32(                        \
            false, amat, false, bmat, (short)0, czero, false, false);         \
        const int tbase = (T) * 16 + rowoff;                                  \
        _Pragma("unroll")                                                     \
        for (int r = 0; r < 8; ++r) {                                         \
            float sc = c[r];                                                  \
            if (sc < vk[0]) {                                                 \
                vk[0] = sc; ik[0] = tbase + r;                                \
                _Pragma("unroll")                                             \
                for (int j = 0; j < KNN - 1; ++j) {                           \
                    bool  sw = vk[j] < vk[j + 1];                             \
                    float tv = vk[j];  int ti = ik[j];                        \
                    vk[j]     = sw ? vk[j + 1] : vk[j];                       \
                    ik[j]     = sw ? ik[j + 1] : ik[j];                       \
                    vk[j + 1] = sw ? tv        : vk[j + 1];                   \
                    ik[j + 1] = sw ? ti        : ik[j + 1];                   \
                }                                                             \
            }                                                                 \
        }                                                                     \
    }

    // two-tile-deep software pipeline: loads for t+2/t+3 issued before the
    // WMMA + scan of t/t+1, hiding L2 latency (prefetch index clamped, no OOB)
    v4f p0 = tp[0 * 16 + lid];
    v4f p1 = tp[1 * 16 + lid];
    for (int t = 0; t < NTILES; t += 2) {
        const int tn = (t + 2 < NTILES) ? (t + 2) : (NTILES - 2);
        v4f n0 = tp[(tn + 0) * 16 + lid];
        v4f n1 = tp[(tn + 1) * 16 + lid];
        PROCESS_TILE(p0, t);
        PROCESS_TILE(p1, t + 1);
        p0 = n0;
        p1 = n1;
    }
#undef PROCESS_TILE

    // ---- merge half-wave partner lists (lanes<16 end with the final top-16) ----
#pragma unroll
    for (int j = 0; j < KNN; ++j) {
        float pv = __shfl(vk[j], lane ^ 16, 32);       // full EXEC at shuffle
        int   pi = __shfl(ik[j], lane ^ 16, 32);
        if (halfsel == 0 && pv < vk[0]) {
            vk[0] = pv; ik[0] = pi;
#pragma unroll
            for (int j2 = 0; j2 < KNN - 1; ++j2) {
                bool  sw = vk[j2] < vk[j2 + 1];
                float tv = vk[j2];  int ti = ik[j2];
                vk[j2]     = sw ? vk[j2 + 1] : vk[j2];
                ik[j2]     = sw ? ik[j2 + 1] : ik[j2];
                vk[j2 + 1] = sw ? tv         : vk[j2 + 1];
                ik[j2 + 1] = sw ? ti         : ik[j2 + 1];
            }
        }
    }

    // ---- normalized Gaussian weights (valid on lanes 0-15) ----
    float w[KNN];
    float wsum = 0.0f;
#pragma unroll
    for (int j = 0; j < KNN; ++j) {
        float d2 = fmaxf(vk[j] + snorm, 0.0f);         // true squared distance
        w[j] = __expf(-0.5f * d2);                     // SIGMA = 1
        wsum += w[j];
    }
    float winv = 1.0f / wsum;
#pragma unroll
    for (int j = 0; j < KNN; ++j) w[j] *= winv;

    // ---- aggregation: broadcast (w, idx) from owning lane, all 32 lanes
    //      cover C=128 with one float4 each (coalesced b128 load/store) ----
    const float* tf = tfeat + (size_t)batch * NA_ * C_;
    for (int s = 0; s < 16; ++s) {
        float a0 = 0.f, a1 = 0.f, a2 = 0.f, a3 = 0.f;
#pragma unroll
        for (int k = 0; k < KNN; ++k) {
            float wk = __shfl(w[k],  s, 32);
            int   id = __shfl(ik[k], s, 32);
            const v4f f = *(const v4f*)(tf + (size_t)id * C_ + lane * 4);
            a0 += wk * f.x; a1 += wk * f.y; a2 += wk * f.z; a3 += wk * f.w;
        }
        v4f acc; acc.x = a0; acc.y = a1; acc.z = a2; acc.w = a3;
        float* op = out + ((size_t)batch * NB_ + tile * 16 + s) * C_ + lane * 4;
        *(v4f*)op = acc;
    }
}

// ---------------------------------------------------------------------------
extern "C" void kernel_launch(void* const* d_in, const int* in_sizes, int n_in,
                              void* d_out, int out_size, void* d_ws, size_t ws_size,
                              hipStream_t stream)
{
    (void)in_sizes; (void)n_in; (void)out_size; (void)ws_size;
    const float* tfeat = (const float*)d_in[0];   // [B, NA, C] f32
    const float* tpos  = (const float*)d_in[1];   // [B, NA, 3] f32
    const float* spos  = (const float*)d_in[2];   // [B, NB, 3] f32
    float*       out   = (float*)d_out;           // [B, NB, C] f32
    v4f*         tpack = (v4f*)d_ws;              // B*NA*16 bytes = 512 KB

    const int total = B_ * NA_;
    pack_teachers_kernel<<<(total + 255) / 256, 256, 0, stream>>>(tpos, tpack, total);

    // one wave32 per 16-student tile; single-wave workgroups spread across WGPs
    knn_agg_kernel<<<B_ * (NB_ / 16), 32, 0, stream>>>(tfeat, tpack, spos, out);
}